// S4SSM_21380347200324
// MI455X (gfx1250) — compile-verified
//
#include <hip/hip_runtime.h>
#include <hip/hip_bf16.h>
#include <math.h>

// ---------------------------------------------------------------------------
// S4 (DPLR SSM) forward for MI455X / gfx1250, f32 end-to-end.
//   1) setup:  discretize; pre-swizzle WMMA A-operands; build rank-1 unroll
//              constants g_j = Q^H Lbar^j Pbar and extra GEMM rows (Q^H Lbar^j Bbar)
//   2) gemm1:  Bu (128 rows) + qb streams (6 extra rows)  via v_wmma_f32_16x16x4_f32
//   3) scan:   4-step-unrolled DPLR recurrence, 1 butterfly per 4 steps
//   4) gemm2:  y = [Cr | -Ci] @ xs_interleaved + D*u      via v_wmma_f32_16x16x4_f32
// ---------------------------------------------------------------------------

typedef float v2f __attribute__((ext_vector_type(2)));
typedef float v8f __attribute__((ext_vector_type(8)));

#define S4_H      128
#define S4_N      64
#define S4_BATCH  16
#define S4_L      8192
#define S4_TOKENS (S4_BATCH * S4_L)   // 131072
#define XPAD      140                 // LDS row stride (floats): 140 % 64 = 12 -> 16 distinct banks

// workspace layout (float offsets)
#define WS_PARAMS 0                          // 576 floats (8*64 vectors + g0..g2 at +512)
#define WS_W1OPS  576                        // 9 mtiles * 2048 = 18432 floats
#define WS_W2OPS  (576 + 18432)              // 16384 floats
#define WS_QB     (576 + 18432 + 16384)      // TOKENS * 8 floats (qb0,qb1,qb2 complex + pad)
#define WS_BU     (WS_QB + S4_TOKENS * 8)
#define WS_XS     (WS_BU + S4_TOKENS * 128)

struct cf { float r, i; };
__device__ __forceinline__ cf cadd(cf a, cf b) { return { a.r + b.r, a.i + b.i }; }
__device__ __forceinline__ cf csub(cf a, cf b) { return { a.r - b.r, a.i - b.i }; }
__device__ __forceinline__ cf cmul(cf a, cf b) { return { a.r * b.r - a.i * b.i,
                                                          a.r * b.i + a.i * b.r }; }

// ---------------------------------------------------------------------------
// Setup.  Wops layout: Wops[((mt*32+kb)*32+lane)*2 + c] =
//   W[mt*16 + (lane&15)][4*kb + 2*(lane>>4) + c]  -> one coalesced b64 per wmma.
// W1 rows 0..127: interleaved Re/Im of Bbar.  Rows 128..133: Re/Im of
// (Q^H Lbar^j Bbar), j=0..2 (qb streams).  Rows 134..143: zero.
// ---------------------------------------------------------------------------
__global__ __launch_bounds__(128)
void s4_setup(const float* __restrict__ llr, const float* __restrict__ li,
              const float* __restrict__ Pr,  const float* __restrict__ Pi,
              const float* __restrict__ Qr,  const float* __restrict__ Qi,
              const float* __restrict__ Br,  const float* __restrict__ Bi,
              const float* __restrict__ Cr,  const float* __restrict__ Ci,
              const float* __restrict__ logdelta, float* __restrict__ ws) {
  __shared__ float s_lbr[64], s_lbi[64], s_pbr[64], s_pbi[64],
                   s_qr[64],  s_qi[64],  s_fr[64],  s_fi[64];
  const int t = threadIdx.x;
  const float delta = expf(logdelta[0]);

  if (t < S4_N) {
    const float lr  = -expf(llr[t]);
    const float liv = li[t];
    const float Ldr = lr * delta, Ldi = liv * delta;
    const float e   = expf(Ldr);
    const float lbr = e * cosf(Ldi);        // Lambda_bar
    const float lbi = e * sinf(Ldi);
    float fr, fi;
    const float mag = sqrtf(Ldr * Ldr + Ldi * Ldi);
    if (mag < 1e-4f) {
      fr = delta * (1.f + 0.5f * Ldr + (Ldr * Ldr - Ldi * Ldi) * (1.f / 6.f));
      fi = delta * (0.5f * Ldi + (2.f * Ldr * Ldi) * (1.f / 6.f));
    } else {
      const float nr = lbr - 1.f, ni = lbi;
      const float dr = lr + 1e-12f, di = liv + 1e-12f;
      const float den = dr * dr + di * di;
      fr = (nr * dr + ni * di) / den;
      fi = (ni * dr - nr * di) / den;
    }
    const float pbr = fr * Pr[t] - fi * Pi[t];
    const float pbi = fr * Pi[t] + fi * Pr[t];
    s_lbr[t] = lbr; s_lbi[t] = lbi; s_fr[t] = fr; s_fi[t] = fi;
    s_pbr[t] = pbr; s_pbi[t] = pbi; s_qr[t] = Qr[t]; s_qi[t] = Qi[t];
    ws[WS_PARAMS + 0 * 64 + t] = lbr;   // Lambda_bar
    ws[WS_PARAMS + 1 * 64 + t] = lbi;
    ws[WS_PARAMS + 2 * 64 + t] = pbr;   // P_bar
    ws[WS_PARAMS + 3 * 64 + t] = pbi;
    ws[WS_PARAMS + 4 * 64 + t] = Qr[t]; // raw Q (conj applied downstream)
    ws[WS_PARAMS + 5 * 64 + t] = Qi[t];
    ws[WS_PARAMS + 6 * 64 + t] = fr;
    ws[WS_PARAMS + 7 * 64 + t] = fi;
  }
  __syncthreads();

  // g_j = Q^H Lbar^j Pbar, j = 0..2
  if (t < 3) {
    float ar = 0.f, ai = 0.f;
    for (int n = 0; n < 64; ++n) {
      cf c = { s_qr[n], -s_qi[n] };
      const cf lam = { s_lbr[n], s_lbi[n] };
      for (int p = 0; p < t; ++p) c = cmul(c, lam);
      const cf g = cmul(c, (cf){ s_pbr[n], s_pbi[n] });
      ar += g.r; ai += g.i;
    }
    ws[WS_PARAMS + 512 + 2 * t]     = ar;
    ws[WS_PARAMS + 512 + 2 * t + 1] = ai;
  }

  for (int idx = t; idx < 18432; idx += 128) {
    const int c    = idx & 1;
    const int lane = (idx >> 1) & 31;
    const int kb   = (idx >> 6) & 31;
    const int mt   = idx >> 11;
    const int m    = mt * 16 + (lane & 15);
    const int k    = 4 * kb + 2 * (lane >> 4) + c;
    float w1;
    if (m < 128) {                       // rows of B_bar (Re/Im interleaved)
      const int n = m >> 1;
      const float fr = s_fr[n], fi = s_fi[n];
      const float br = Br[n * S4_H + k], bi = Bi[n * S4_H + k];
      w1 = (m & 1) ? (fr * bi + fi * br) : (fr * br - fi * bi);
    } else {                             // qb rows: (Q^H Lbar^j Bbar)[h]
      const int rr = m - 128, jj = rr >> 1, ss = rr & 1;
      float vr = 0.f, vi = 0.f;
      if (jj < 3) {
        for (int n = 0; n < 64; ++n) {
          cf cc = { s_qr[n], -s_qi[n] };
          const cf lam = { s_lbr[n], s_lbi[n] };
          for (int p = 0; p < jj; ++p) cc = cmul(cc, lam);
          cc = cmul(cc, (cf){ s_fr[n], s_fi[n] });
          const cf v = cmul(cc, (cf){ Br[n * S4_H + k], Bi[n * S4_H + k] });
          vr += v.r; vi += v.i;
        }
      }
      w1 = ss ? vi : vr;
    }
    ws[WS_W1OPS + idx] = w1;
    if (idx < 16384) {                   // W2: y = Cr*xr - Ci*xi, xs interleaved
      const int p = k >> 1;
      ws[WS_W2OPS + idx] = (k & 1) ? -Ci[m * S4_N + p] : Cr[m * S4_N + p];
    }
  }
}

// ---------------------------------------------------------------------------
// GEMM: OUT[token][0..127] = W(128..144 rows) @ X[token][0..127].
// NMT=9 + WRITEQB: rows 128..133 harvested to QB[token][0..7].
// ADDDU: fused OUT += D[h]*U[token][h].
// ---------------------------------------------------------------------------
template <int NMT, bool ADDDU, bool WRITEQB>
__global__ __launch_bounds__(128)
void s4_gemm_wmma(const float* __restrict__ X, const float* __restrict__ Wops,
                  float* __restrict__ OUT, const float* __restrict__ Dvec,
                  const float* __restrict__ U, float* __restrict__ QB) {
  __shared__ float Xl[4][16 * XPAD];   // per-wave activation tile, reused as out tile
  const int lane = threadIdx.x & 31;
  const int wave = threadIdx.x >> 5;
  const int tt   = (blockIdx.x * 4 + wave) * 16;   // token tile base
  float* xl = Xl[wave];

  const float4* X4 = (const float4*)X;
  for (int j = 0; j < 16; ++j) {
    const int fidx = j * 32 + lane;
    const int row = fidx >> 5, q = fidx & 31;
    const float4 v = X4[(size_t)(tt + row) * 32 + q];
    *(float4*)&xl[row * XPAD + q * 4] = v;
  }
  __syncthreads();

  v8f acc[NMT] = {};
  const v2f* Wv = (const v2f*)Wops;
  const int hi = lane >> 4;           // K half split across half-waves
  const int ln = lane & 15;

  for (int kb = 0; kb < 32; ++kb) {
    const int k0 = kb * 4;
    const v2f b = *(const v2f*)&xl[ln * XPAD + k0 + 2 * hi];
    #pragma unroll
    for (int mt = 0; mt < NMT; ++mt) {
      const v2f a = Wv[(mt * 32 + kb) * 32 + lane];
      acc[mt] = __builtin_amdgcn_wmma_f32_16x16x4_f32(
          false, a, false, b, (short)0, acc[mt], false, false);
    }
  }
  __syncthreads();

  #pragma unroll
  for (int mt = 0; mt < NMT; ++mt) {
    if (mt == 8 && hi) continue;       // rows 136..143 are zero padding
    #pragma unroll
    for (int r = 0; r < 8; ++r)
      xl[ln * XPAD + mt * 16 + r + hi * 8] = acc[mt][r];
  }
  __syncthreads();

  const float4* U4 = (const float4*)U;
  const float4* D4 = (const float4*)Dvec;
  float4* O4 = (float4*)OUT;
  for (int j = 0; j < 16; ++j) {
    const int fidx = j * 32 + lane;
    const int row = fidx >> 5, q = fidx & 31;
    float4 y = *(const float4*)&xl[row * XPAD + q * 4];
    if (ADDDU) {
      const float4 uv = U4[(size_t)(tt + row) * 32 + q];
      const float4 dv = D4[q];
      y.x += dv.x * uv.x; y.y += dv.y * uv.y;
      y.z += dv.z * uv.z; y.w += dv.w * uv.w;
    }
    O4[(size_t)(tt + row) * 32 + q] = y;
  }
  if (WRITEQB) {                        // cols 128..135 -> QB[token][8]
    const int tok = lane >> 1, part = lane & 1;
    const float4 v = *(const float4*)&xl[tok * XPAD + 128 + part * 4];
    ((float4*)QB)[(size_t)(tt + tok) * 2 + part] = v;
  }
}

// ---------------------------------------------------------------------------
// 4-step-unrolled scan.  One wave per batch; 2 complex states per lane.
// Per 4 steps: one 5-stage butterfly over 8 packed floats r_j = (Q^H Lbar^j) x,
// then scalar complex algebra for s1..s3 and 4 cheap elementwise updates.
// ---------------------------------------------------------------------------
__global__ __launch_bounds__(32)
void s4_scan4(const float* __restrict__ prm, const float* __restrict__ bu_all,
              const float* __restrict__ qb_all, float* __restrict__ xs_all) {
  const int b = blockIdx.x;
  const int i = threadIdx.x;           // states i and i+32
  const int j = i + 32;
  const cf l0 = { prm[0 * 64 + i], prm[1 * 64 + i] };
  const cf l1 = { prm[0 * 64 + j], prm[1 * 64 + j] };
  const cf p0 = { prm[2 * 64 + i], prm[3 * 64 + i] };
  const cf p1 = { prm[2 * 64 + j], prm[3 * 64 + j] };
  const cf q0 = { prm[4 * 64 + i], -prm[5 * 64 + i] };   // conj(Q)
  const cf q1 = { prm[4 * 64 + j], -prm[5 * 64 + j] };
  const cf g0 = { prm[512], prm[513] };
  const cf g1 = { prm[514], prm[515] };
  const cf g2 = { prm[516], prm[517] };
  const cf w0a = q0,            w0b = q1;
  const cf w1a = cmul(q0, l0),  w1b = cmul(q1, l1);
  const cf w2a = cmul(w1a, l0), w2b = cmul(w1b, l1);
  const cf w3a = cmul(w2a, l0), w3b = cmul(w2b, l1);

  const float* bu = bu_all + (size_t)b * S4_L * 128;
  const float* qb = qb_all + (size_t)b * S4_L * 8;
  float*       xs = xs_all + (size_t)b * S4_L * 128;

  cf x0 = { 0.f, 0.f }, x1 = { 0.f, 0.f };

  float2 cb0[4], cb1[4];
  #pragma unroll
  for (int s = 0; s < 4; ++s) {
    cb0[s] = *(const float2*)(bu + (size_t)s * 128 + 2 * i);
    cb1[s] = *(const float2*)(bu + (size_t)s * 128 + 2 * j);
  }
  float4 cqA  = *(const float4*)(qb + 0);
  float2 cqA2 = *(const float2*)(qb + 4);
  float4 cqB  = *(const float4*)(qb + 8);
  float2 cqC  = *(const float2*)(qb + 16);

  for (int t = 0; t < S4_L; t += 4) {
    const int tn = (t + 4 < S4_L) ? t + 4 : t;
    float2 nb0[4], nb1[4];
    #pragma unroll
    for (int s = 0; s < 4; ++s) {
      nb0[s] = *(const float2*)(bu + (size_t)(tn + s) * 128 + 2 * i);
      nb1[s] = *(const float2*)(bu + (size_t)(tn + s) * 128 + 2 * j);
    }
    const float4 nqA  = *(const float4*)(qb + (size_t)tn * 8);
    const float2 nqA2 = *(const float2*)(qb + (size_t)tn * 8 + 4);
    const float4 nqB  = *(const float4*)(qb + (size_t)(tn + 1) * 8);
    const float2 nqC  = *(const float2*)(qb + (size_t)(tn + 2) * 8);
    const int tp = (t + 16 < S4_L) ? t + 16 : t;
    __builtin_prefetch(bu + (size_t)tp * 128 + 2 * i, 0, 1);

    // packed reduction partials r_j = (Q^H Lbar^j) x
    const cf r0 = cadd(cmul(w0a, x0), cmul(w0b, x1));
    const cf r1 = cadd(cmul(w1a, x0), cmul(w1b, x1));
    const cf r2 = cadd(cmul(w2a, x0), cmul(w2b, x1));
    const cf r3 = cadd(cmul(w3a, x0), cmul(w3b, x1));
    float v0 = r0.r, v1 = r0.i, v2 = r1.r, v3 = r1.i;
    float v4 = r2.r, v5 = r2.i, v6 = r3.r, v7 = r3.i;
    #pragma unroll
    for (int off = 16; off > 0; off >>= 1) {
      v0 += __shfl_xor(v0, off, 32); v1 += __shfl_xor(v1, off, 32);
      v2 += __shfl_xor(v2, off, 32); v3 += __shfl_xor(v3, off, 32);
      v4 += __shfl_xor(v4, off, 32); v5 += __shfl_xor(v5, off, 32);
      v6 += __shfl_xor(v6, off, 32); v7 += __shfl_xor(v7, off, 32);
    }

    // scalar chain: s_t..s_{t+3}
    const cf qb0t  = { cqA.x, cqA.y }, qb1t  = { cqA.z, cqA.w }, qb2t = { cqA2.x, cqA2.y };
    const cf qb0t1 = { cqB.x, cqB.y }, qb1t1 = { cqB.z, cqB.w };
    const cf qb0t2 = { cqC.x, cqC.y };
    const cf s0 = { v0, v1 };
    const cf s1 = cadd(csub((cf){ v2, v3 }, cmul(g0, s0)), qb0t);
    const cf s2 = cadd(csub(cadd(csub((cf){ v4, v5 }, cmul(g1, s0)), qb1t),
                            cmul(g0, s1)), qb0t1);
    const cf s3 = cadd(csub(cadd(csub(cadd(csub((cf){ v6, v7 }, cmul(g2, s0)), qb2t),
                                      cmul(g1, s1)), qb1t1),
                            cmul(g0, s2)), qb0t2);

    // four elementwise updates:  x = Lbar*x - s*Pbar + b
    const cf ss[4] = { s0, s1, s2, s3 };
    #pragma unroll
    for (int s = 0; s < 4; ++s) {
      const cf n0 = { l0.r * x0.r - l0.i * x0.i - (ss[s].r * p0.r - ss[s].i * p0.i) + cb0[s].x,
                      l0.r * x0.i + l0.i * x0.r - (ss[s].r * p0.i + ss[s].i * p0.r) + cb0[s].y };
      const cf n1 = { l1.r * x1.r - l1.i * x1.i - (ss[s].r * p1.r - ss[s].i * p1.i) + cb1[s].x,
                      l1.r * x1.i + l1.i * x1.r - (ss[s].r * p1.i + ss[s].i * p1.r) + cb1[s].y };
      x0 = n0; x1 = n1;
      *(float2*)(xs + (size_t)(t + s) * 128 + 2 * i) = make_float2(x0.r, x0.i);
      *(float2*)(xs + (size_t)(t + s) * 128 + 2 * j) = make_float2(x1.r, x1.i);
    }

    #pragma unroll
    for (int s = 0; s < 4; ++s) { cb0[s] = nb0[s]; cb1[s] = nb1[s]; }
    cqA = nqA; cqA2 = nqA2; cqB = nqB; cqC = nqC;
  }
}

// ---------------------------------------------------------------------------
extern "C" void kernel_launch(void* const* d_in, const int* in_sizes, int n_in,
                              void* d_out, int out_size, void* d_ws, size_t ws_size,
                              hipStream_t stream) {
  (void)in_sizes; (void)n_in; (void)out_size; (void)ws_size;
  const float* u   = (const float*)d_in[0];
  const float* llr = (const float*)d_in[1];
  const float* li  = (const float*)d_in[2];
  const float* Pr  = (const float*)d_in[3];
  const float* Pi  = (const float*)d_in[4];
  const float* Qr  = (const float*)d_in[5];
  const float* Qi  = (const float*)d_in[6];
  const float* Br  = (const float*)d_in[7];
  const float* Bi  = (const float*)d_in[8];
  const float* Cr  = (const float*)d_in[9];
  const float* Ci  = (const float*)d_in[10];
  const float* D   = (const float*)d_in[11];
  const float* ld  = (const float*)d_in[12];
  float* ws = (float*)d_ws;
  float* y  = (float*)d_out;

  s4_setup<<<1, 128, 0, stream>>>(llr, li, Pr, Pi, Qr, Qi, Br, Bi, Cr, Ci, ld, ws);
  s4_gemm_wmma<9, false, true><<<S4_TOKENS / 64, 128, 0, stream>>>(
      u, ws + WS_W1OPS, ws + WS_BU, nullptr, nullptr, ws + WS_QB);
  s4_scan4<<<S4_BATCH, 32, 0, stream>>>(ws + WS_PARAMS, ws + WS_BU,
                                        ws + WS_QB, ws + WS_XS);
  s4_gemm_wmma<8, true, false><<<S4_TOKENS / 64, 128, 0, stream>>>(
      ws + WS_XS, ws + WS_W2OPS, y, D, u, nullptr);
}